// SC2_PCR_9388798509735
// MI455X (gfx1250) — compile-verified
//
#include <hip/hip_runtime.h>
#include <hip/hip_bf16.h>

typedef float v2f __attribute__((ext_vector_type(2)));
typedef float v8f __attribute__((ext_vector_type(8)));

#define N_PTS 6144
#define D_THRE_INV 10.0f   // 1 / 0.1
#define NUM_ITERS 10
#define ROWS_PER_BLK 16
#define JCHUNK 512
#define KWAVES 4           // waves per block in the WMMA matvec (K split 4 ways)

// ---------------------------------------------------------------------------
// Phase 1: materialize SC row-major fp32.
// SC[i,j] = max(0, 1 - ((|d2(i,j)-d3(i,j)|)/0.1)^2)
// Block handles 16 rows; j-points staged through LDS in chunks for reuse.
// ---------------------------------------------------------------------------
__global__ void __launch_bounds__(256) build_sc(const float* __restrict__ p2,
                                                const float* __restrict__ p3,
                                                float* __restrict__ SC) {
  __shared__ float s2x[JCHUNK], s2y[JCHUNK];
  __shared__ float s3x[JCHUNK], s3y[JCHUNK], s3z[JCHUNK];
  const int i0  = blockIdx.x * ROWS_PER_BLK;
  const int tid = threadIdx.x;

  for (int jc = 0; jc < N_PTS; jc += JCHUNK) {
    __syncthreads();
    for (int t = tid; t < JCHUNK; t += 256) {
      int j = jc + t;
      s2x[t] = p2[2 * j];
      s2y[t] = p2[2 * j + 1];
      s3x[t] = p3[3 * j];
      s3y[t] = p3[3 * j + 1];
      s3z[t] = p3[3 * j + 2];
    }
    __syncthreads();
    for (int r = 0; r < ROWS_PER_BLK; ++r) {
      const int i = i0 + r;
      const float x2 = p2[2 * i], y2 = p2[2 * i + 1];
      const float x3 = p3[3 * i], y3 = p3[3 * i + 1], z3 = p3[3 * i + 2];
      const int jj = tid * 2;  // two consecutive j per thread -> float2 store
      float2 o;
      float* op = reinterpret_cast<float*>(&o);
#pragma unroll
      for (int u = 0; u < 2; ++u) {
        const int t = jj + u;
        float dx2 = x2 - s2x[t], dy2 = y2 - s2y[t];
        float d2 = sqrtf(dx2 * dx2 + dy2 * dy2);
        float dx3 = x3 - s3x[t], dy3 = y3 - s3y[t], dz3 = z3 - s3z[t];
        float d3 = sqrtf(dx3 * dx3 + dy3 * dy3 + dz3 * dz3);
        float c = fabsf(d2 - d3) * D_THRE_INV;
        float s = 1.0f - c * c;
        op[u] = s > 0.0f ? s : 0.0f;
      }
      *reinterpret_cast<float2*>(SC + (size_t)i * N_PTS + jc + jj) = o;
    }
  }
}

// ---------------------------------------------------------------------------
// v0 = ones
// ---------------------------------------------------------------------------
__global__ void init_ones(float* __restrict__ v) {
  int i = blockIdx.x * blockDim.x + threadIdx.x;
  if (i < N_PTS) v[i] = 1.0f;
}

// ---------------------------------------------------------------------------
// Phase 2: y = SC * v  via V_WMMA_F32_16X16X4_F32.
// Block = 4 waves, each wave owns one 16-row M-tile and 1/4 of K.
// fp32 A-fragment (16x4): lane<16 -> (M=lane, K=k..k+1), lane>=16 -> (M=lane-16,
// K=k+2..k+3); with row-major SC that is one b64 load per lane.
// B-fragment (4x16) carries v in column 0 only (lanes 0 and 16).
// Column 0 of D: lane 0 VGPR g = y[m0+g], lane 16 VGPR g = y[m0+8+g].
// ---------------------------------------------------------------------------
__global__ void __launch_bounds__(32 * KWAVES) matvec_wmma(
    const float* __restrict__ SC, const float* __restrict__ v,
    float* __restrict__ y) {
  __shared__ float lv[N_PTS];            // 24 KB staged vector
  __shared__ float red[KWAVES][32][8];   // 4 KB partial-C reduction
  const int tid  = threadIdx.x;
  const int lane = tid & 31;
  const int wave = tid >> 5;

  for (int t = tid; t < N_PTS; t += 32 * KWAVES) lv[t] = v[t];
  __syncthreads();

  const int m0    = blockIdx.x * 16;
  const int kbase = wave * (N_PTS / KWAVES);           // 1536 K per wave
  const int mrow  = m0 + (lane & 15);
  const int ksel  = (lane < 16) ? 0 : 2;
  const float bmask = ((lane & 15) == 0) ? 1.0f : 0.0f; // lanes 0,16 = column 0
  const float* aptr = SC + (size_t)mrow * N_PTS + kbase + ksel;

  v8f acc0 = {}, acc1 = {}, acc2 = {}, acc3 = {};
  for (int k = 0; k < N_PTS / KWAVES; k += 16) {
    v2f a0 = *reinterpret_cast<const v2f*>(aptr + k);
    v2f a1 = *reinterpret_cast<const v2f*>(aptr + k + 4);
    v2f a2 = *reinterpret_cast<const v2f*>(aptr + k + 8);
    v2f a3 = *reinterpret_cast<const v2f*>(aptr + k + 12);
    const int kb = kbase + k + ksel;
    v2f b0 = {bmask * lv[kb],      bmask * lv[kb + 1]};
    v2f b1 = {bmask * lv[kb + 4],  bmask * lv[kb + 5]};
    v2f b2 = {bmask * lv[kb + 8],  bmask * lv[kb + 9]};
    v2f b3 = {bmask * lv[kb + 12], bmask * lv[kb + 13]};
    acc0 = __builtin_amdgcn_wmma_f32_16x16x4_f32(false, a0, false, b0,
                                                 (short)0, acc0, false, false);
    acc1 = __builtin_amdgcn_wmma_f32_16x16x4_f32(false, a1, false, b1,
                                                 (short)0, acc1, false, false);
    acc2 = __builtin_amdgcn_wmma_f32_16x16x4_f32(false, a2, false, b2,
                                                 (short)0, acc2, false, false);
    acc3 = __builtin_amdgcn_wmma_f32_16x16x4_f32(false, a3, false, b3,
                                                 (short)0, acc3, false, false);
  }
  v8f c = acc0 + acc1 + acc2 + acc3;
#pragma unroll
  for (int g = 0; g < 8; ++g) red[wave][lane][g] = c[g];
  __syncthreads();

  if (wave == 0) {
#pragma unroll
    for (int g = 0; g < 8; ++g)
      c[g] = red[0][lane][g] + red[1][lane][g] + red[2][lane][g] + red[3][lane][g];
    if ((lane & 15) == 0) {
      float* outp = y + m0 + ((lane == 16) ? 8 : 0);
#pragma unroll
      for (int g = 0; g < 8; ++g) outp[g] = c[g];
    }
  }
}

// ---------------------------------------------------------------------------
// Fallback matvec (used only if workspace cannot hold the 151 MB matrix):
// recompute SC(i,j) on the fly. One block per output row, fixed-order
// reduction (deterministic).
// ---------------------------------------------------------------------------
__global__ void __launch_bounds__(256) matvec_fused(
    const float* __restrict__ p2, const float* __restrict__ p3,
    const float* __restrict__ v, float* __restrict__ y) {
  __shared__ float part[256];
  const int i = blockIdx.x;
  const float x2 = p2[2 * i], y2 = p2[2 * i + 1];
  const float x3 = p3[3 * i], y3 = p3[3 * i + 1], z3 = p3[3 * i + 2];
  float s = 0.0f;
  for (int j = threadIdx.x; j < N_PTS; j += 256) {
    float dx2 = x2 - p2[2 * j], dy2 = y2 - p2[2 * j + 1];
    float d2 = sqrtf(dx2 * dx2 + dy2 * dy2);
    float dx3 = x3 - p3[3 * j], dy3 = y3 - p3[3 * j + 1], dz3 = z3 - p3[3 * j + 2];
    float d3 = sqrtf(dx3 * dx3 + dy3 * dy3 + dz3 * dz3);
    float c = fabsf(d2 - d3) * D_THRE_INV;
    float sc = 1.0f - c * c;
    sc = sc > 0.0f ? sc : 0.0f;
    s += sc * v[j];
  }
  part[threadIdx.x] = s;
  __syncthreads();
  for (int off = 128; off > 0; off >>= 1) {
    if (threadIdx.x < off) part[threadIdx.x] += part[threadIdx.x + off];
    __syncthreads();
  }
  if (threadIdx.x == 0) y[i] = part[0];
}

// ---------------------------------------------------------------------------
// v = y / (||y|| + 1e-6), single block.
// ---------------------------------------------------------------------------
__global__ void __launch_bounds__(1024) normalize_k(const float* __restrict__ y,
                                                    float* __restrict__ out) {
  __shared__ float part[1024];
  float s = 0.0f;
  for (int i = threadIdx.x; i < N_PTS; i += 1024) {
    float t = y[i];
    s += t * t;
  }
  part[threadIdx.x] = s;
  __syncthreads();
  for (int off = 512; off > 0; off >>= 1) {
    if (threadIdx.x < off) part[threadIdx.x] += part[threadIdx.x + off];
    __syncthreads();
  }
  const float inv = 1.0f / (sqrtf(part[0]) + 1e-6f);
  for (int i = threadIdx.x; i < N_PTS; i += 1024) out[i] = y[i] * inv;
}

// ---------------------------------------------------------------------------
extern "C" void kernel_launch(void* const* d_in, const int* in_sizes, int n_in,
                              void* d_out, int out_size, void* d_ws,
                              size_t ws_size, hipStream_t stream) {
  (void)in_sizes; (void)n_in; (void)out_size;
  const float* p2 = (const float*)d_in[0];  // 6144 x 2 f32
  const float* p3 = (const float*)d_in[1];  // 6144 x 3 f32
  float* out = (float*)d_out;               // 6144 f32 (confidence)
  char* ws = (char*)d_ws;

  const size_t sc_bytes = (size_t)N_PTS * N_PTS * sizeof(float);  // ~151 MB
  const size_t vec_bytes = (size_t)N_PTS * sizeof(float);

  if (ws_size >= sc_bytes + 2 * vec_bytes) {
    // Materialized path: SC lives in L2 (192 MB), matvec via f32 WMMA.
    float* SC = (float*)ws;
    float* v  = (float*)(ws + sc_bytes);
    float* y  = v + N_PTS;
    build_sc<<<N_PTS / ROWS_PER_BLK, 256, 0, stream>>>(p2, p3, SC);
    init_ones<<<(N_PTS + 255) / 256, 256, 0, stream>>>(v);
    for (int it = 0; it < NUM_ITERS; ++it) {
      matvec_wmma<<<N_PTS / 16, 32 * KWAVES, 0, stream>>>(SC, v, y);
      normalize_k<<<1, 1024, 0, stream>>>(y, (it == NUM_ITERS - 1) ? out : v);
    }
  } else {
    // Fallback: recompute SC on the fly each iteration.
    float* v = (float*)ws;
    float* y = v + N_PTS;
    init_ones<<<(N_PTS + 255) / 256, 256, 0, stream>>>(v);
    for (int it = 0; it < NUM_ITERS; ++it) {
      matvec_fused<<<N_PTS, 256, 0, stream>>>(p2, p3, v, y);
      normalize_k<<<1, 1024, 0, stream>>>(y, (it == NUM_ITERS - 1) ? out : v);
    }
  }
}